// GMMGCNN_39049842655441
// MI455X (gfx1250) — compile-verified
//
#include <hip/hip_runtime.h>
#include <math.h>

// ---------------------------------------------------------------------------
// GMM-imputed GCNN forward for MI455X (gfx1250, wave32, WMMA).
// All matrix math runs through V_WMMA_F32_16X16X4_F32 (fp32 in/out, K-step 4).
// Wave tile 32x64 (two A row-groups per B fragment); __launch_bounds__(256,1)
// keeps the 128 accumulator VGPRs resident (no scratch spills in the k-loop).
// ---------------------------------------------------------------------------

typedef float v2f __attribute__((ext_vector_type(2)));
typedef float v8f __attribute__((ext_vector_type(8)));

#define WB   32    // batch
#define NN   500   // nodes
#define NP   512   // padded nodes
#define KK   5     // mixture components
#define F0D  144
#define H0D  256
#define H1D  256
#define PPD  288
#define LN2PI 1.8378770664093453f
#define INV_SQRT_2PI 0.3989422804014327f
#define INV_SQRT_2   0.7071067811865475f

// ---- WMMA f32 16x16x4 fragment helpers (wave32) ---------------------------
// A (16x4): lane L<16 holds row L, K={k,k+1}; lane L>=16 holds row L-16, K={k+2,k+3}
// B (4x16): lane L<16 holds col L, K={k,k+1}; lane L>=16 holds col L-16, K={k+2,k+3}
// C (16x16): VGPR j, lanes 0-15 -> (M=j, N=lane); lanes 16-31 -> (M=8+j, N=lane-16)

static __device__ __forceinline__ v2f ld_a(const float* __restrict__ A, int lda,
                                           int m0, int k, int lane, int M, int Kd) {
  int row = m0 + (lane & 15);
  int kk  = k + ((lane >> 4) << 1);
  v2f a;
  bool ok = row < M;
  a.x = (ok && kk     < Kd) ? A[(size_t)row * lda + kk    ] : 0.f;
  a.y = (ok && kk + 1 < Kd) ? A[(size_t)row * lda + kk + 1] : 0.f;
  return a;
}

static __device__ __forceinline__ v2f ld_b(const float* __restrict__ B, int ldb,
                                           int k, int n0, int lane, int Kd, int Nn) {
  int col = n0 + (lane & 15);
  int kk  = k + ((lane >> 4) << 1);
  v2f b;
  bool ok = col < Nn;
  b.x = (ok && kk     < Kd) ? B[(size_t)(kk    ) * ldb + col] : 0.f;
  b.y = (ok && kk + 1 < Kd) ? B[(size_t)(kk + 1) * ldb + col] : 0.f;
  return b;
}

#define WMMA_F32(a, b, c) \
  __builtin_amdgcn_wmma_f32_16x16x4_f32(false, (a), false, (b), (short)0, (c), false, false)

// E[relu(X)], X ~ N(mu, var)
static __device__ __forceinline__ float ex_relu(float mu, float var) {
  if (var <= 0.f) return fmaxf(mu, 0.f);
  float s = sqrtf(var);
  float w = mu / s;
  return s * (INV_SQRT_2PI * expf(-0.5f * w * w)
              + 0.5f * w * (1.f + erff(w * INV_SQRT_2)));
}

// ---------------------------------------------------------------------------
// Generic batched WMMA GEMM: C[z] = epi(A[z] * B[z]).  Block = 256 thr (8 waves,
// 2 x 4); block tile 64 x 256; wave tile 32 x 64 (2 row-groups x 4 col tiles).
// EPI: 0 = none, 1 = relu, 2 = +bias
// ---------------------------------------------------------------------------
template <int EPI>
__global__ __launch_bounds__(256, 1)
void gemm_wmma_k(const float* __restrict__ A, const float* __restrict__ B,
                 const float* __restrict__ bias, float* __restrict__ C,
                 int M, int Kd, int Nn, int lda, int ldb, int ldc,
                 long long sA, long long sB, long long sC) {
  const int z = blockIdx.z;
  A += (size_t)z * sA;  B += (size_t)z * sB;  C += (size_t)z * sC;
  const int lane = threadIdx.x & 31;
  const int wid  = threadIdx.x >> 5;
  const int m0 = blockIdx.y * 64 + (wid >> 2) * 32;
  const int c0 = blockIdx.x * 256 + (wid & 3) * 64;

  v8f acc[2][4] = {};
  for (int k = 0; k < Kd; k += 4) {
    if (k + 16 < Kd)   // keep next A lines warm (global_prefetch on gfx1250)
      __builtin_prefetch(&A[(size_t)(m0 + (lane & 15)) * lda + k + 16], 0, 1);
    v2f a0 = ld_a(A, lda, m0,      k, lane, M, Kd);
    v2f a1 = ld_a(A, lda, m0 + 16, k, lane, M, Kd);
#pragma unroll
    for (int t = 0; t < 4; ++t) {
      v2f b = ld_b(B, ldb, k, c0 + 16 * t, lane, Kd, Nn);
      acc[0][t] = WMMA_F32(a0, b, acc[0][t]);
      acc[1][t] = WMMA_F32(a1, b, acc[1][t]);
    }
  }
#pragma unroll
  for (int r = 0; r < 2; ++r) {
#pragma unroll
    for (int t = 0; t < 4; ++t) {
      int col = c0 + 16 * t + (lane & 15);
      if (col >= Nn) continue;
      int rbase = m0 + 16 * r + ((lane >> 4) << 3);
#pragma unroll
      for (int j = 0; j < 8; ++j) {
        int row = rbase + j;
        if (row < M) {
          float v = acc[r][t][j];
          if (EPI == 1) v = fmaxf(v, 0.f);
          if (EPI == 2) v += bias[col];
          C[(size_t)row * ldc + col] = v;
        }
      }
    }
  }
}

// ---------------------------------------------------------------------------
// Pad 500x500 -> 512x512 (zero pad)
// ---------------------------------------------------------------------------
__global__ void pad512_k(const float* __restrict__ src, float* __restrict__ dst) {
  int idx = blockIdx.x * 256 + threadIdx.x;
  if (idx >= NP * NP) return;
  int i = idx >> 9, j = idx & (NP - 1);
  dst[idx] = (i < NN && j < NN) ? src[i * NN + j] : 0.f;
}

// Ssum = (I + S + S^2 + S^3)/4 ; Asum = (I + A.A + A^2.A^2 + A^3.A^3)/4
__global__ void combine_k(const float* __restrict__ S1, const float* __restrict__ S2,
                          const float* __restrict__ S3, const float* __restrict__ A1,
                          const float* __restrict__ A2, const float* __restrict__ A3,
                          float* __restrict__ Ssum, float* __restrict__ Asum) {
  int idx = blockIdx.x * 256 + threadIdx.x;
  if (idx >= NP * NP) return;
  int i = idx >> 9, j = idx & (NP - 1);
  float id = (i == j && i < NN) ? 1.f : 0.f;
  Ssum[idx] = 0.25f * (id + S1[idx] + S2[idx] + S3[idx]);
  float a1 = A1[idx], a2 = A2[idx], a3 = A3[idx];
  Asum[idx] = 0.25f * (id + a1 * a1 + a2 * a2 + a3 * a3);
}

// vars/invvars = exp(+/-logvars); cst[k] = log_softmax(logp)[k] - 0.5*(F0*ln2pi + sum logvars[k])
__global__ void prep_k(const float* __restrict__ logvars, const float* __restrict__ logp,
                       float* __restrict__ vars, float* __restrict__ invvars,
                       float* __restrict__ cst) {
  int t = threadIdx.x;
  for (int i = t; i < KK * F0D; i += blockDim.x) {
    float v = expf(logvars[i]);
    vars[i] = v;
    invvars[i] = 1.f / v;
  }
  if (t < KK) {
    float mx = -1e30f;
    for (int k = 0; k < KK; ++k) mx = fmaxf(mx, logp[k]);
    float se = 0.f;
    for (int k = 0; k < KK; ++k) se += expf(logp[k] - mx);
    float lse = mx + logf(se);
    float slv = 0.f;
    for (int f = 0; f < F0D; ++f) slv += logvars[t * F0D + f];
    cst[t] = (logp[t] - lse) - 0.5f * ((float)F0D * LN2PI + slv);
  }
}

// gamma[b,k,n] = softmax_k( cst[k] - 0.5 * sum_{f observed} (x-mu)^2/var )
__global__ void gamma_k(const float* __restrict__ feat, const float* __restrict__ mu,
                        const float* __restrict__ invvars, const float* __restrict__ cst,
                        float* __restrict__ gamma) {
  int idx = blockIdx.x * 256 + threadIdx.x;
  if (idx >= WB * NN) return;
  int b = idx / NN, n = idx % NN;
  float q[KK] = {};
  const float* xr = feat + (size_t)idx * F0D;
  for (int f = 0; f < F0D; ++f) {
    float x = xr[f];
    if (x == x) {  // observed (not NaN)
#pragma unroll
      for (int k = 0; k < KK; ++k) {
        float d = x - mu[k * F0D + f];
        q[k] += d * d * invvars[k * F0D + f];
      }
    }
  }
  float lp[KK], mx = -1e30f;
#pragma unroll
  for (int k = 0; k < KK; ++k) { lp[k] = cst[k] - 0.5f * q[k]; mx = fmaxf(mx, lp[k]); }
  float s = 0.f;
#pragma unroll
  for (int k = 0; k < KK; ++k) { lp[k] = expf(lp[k] - mx); s += lp[k]; }
  float inv = 1.f / s;
#pragma unroll
  for (int k = 0; k < KK; ++k) gamma[((size_t)b * KK + k) * NP + n] = lp[k] * inv;
}

// ---------------------------------------------------------------------------
// Imputed dual GEMM: tx = mean_mat @ W0 ; tc = var_mat @ (W0.W0)
// z = b*K + k ; output [z][512][256] (rows >= 500 are zero)
// Wave tile 32x64; W0 fragments reused across both row-groups and both GEMMs.
// ---------------------------------------------------------------------------
__global__ __launch_bounds__(256, 1)
void impute_gemm_k(const float* __restrict__ feat, const float* __restrict__ mu,
                   const float* __restrict__ vars, const float* __restrict__ W0,
                   float* __restrict__ tx, float* __restrict__ tc) {
  const int z = blockIdx.z;
  const int b = z / KK, k = z % KK;
  const int lane = threadIdx.x & 31;
  const int wid  = threadIdx.x >> 5;
  const int m0 = blockIdx.y * 64 + (wid >> 2) * 32;
  const int c0 = (wid & 3) * 64;

  v8f ax[2][4] = {}, ac[2][4] = {};
  for (int f = 0; f < F0D; f += 4) {
    const int f0 = f + ((lane >> 4) << 1);
    float mu0 = mu[k * F0D + f0],   mu1 = mu[k * F0D + f0 + 1];
    float vv0 = vars[k * F0D + f0], vv1 = vars[k * F0D + f0 + 1];
    v2f am[2], av[2];
#pragma unroll
    for (int r = 0; r < 2; ++r) {
      int n = m0 + 16 * r + (lane & 15);
      am[r].x = 0.f; am[r].y = 0.f; av[r].x = 0.f; av[r].y = 0.f;
      if (n < NN) {
        const float* fr = feat + ((size_t)b * NN + n) * F0D + f0;
        float r0 = fr[0], r1 = fr[1];
        bool nan0 = !(r0 == r0), nan1 = !(r1 == r1);
        am[r].x = nan0 ? mu0 : r0;  am[r].y = nan1 ? mu1 : r1;
        av[r].x = nan0 ? vv0 : 0.f; av[r].y = nan1 ? vv1 : 0.f;
      }
    }
#pragma unroll
    for (int t = 0; t < 4; ++t) {
      v2f bw = ld_b(W0, H0D, f, c0 + 16 * t, lane, F0D, H0D);
      v2f bs; bs.x = bw.x * bw.x; bs.y = bw.y * bw.y;
#pragma unroll
      for (int r = 0; r < 2; ++r) {
        ax[r][t] = WMMA_F32(am[r], bw, ax[r][t]);
        ac[r][t] = WMMA_F32(av[r], bs, ac[r][t]);
      }
    }
  }
  float* txz = tx + (size_t)z * NP * H0D;
  float* tcz = tc + (size_t)z * NP * H0D;
#pragma unroll
  for (int r = 0; r < 2; ++r) {
#pragma unroll
    for (int t = 0; t < 4; ++t) {
      int col = c0 + 16 * t + (lane & 15);
      int rbase = m0 + 16 * r + ((lane >> 4) << 3);
#pragma unroll
      for (int j = 0; j < 8; ++j) {
        int row = rbase + j;
        txz[(size_t)row * H0D + col] = ax[r][t][j];
        tcz[(size_t)row * H0D + col] = ac[r][t][j];
      }
    }
  }
}

// ---------------------------------------------------------------------------
// Fused diffusion + E[relu]: ex = ExRelu( Ssum @ tx[z], Asum @ tc[z] )
// Wave tile 32x64, dual accumulators: 16 wmma per k-step vs 24 loaded dwords.
// ---------------------------------------------------------------------------
__global__ __launch_bounds__(256, 1)
void conv_ex_k(const float* __restrict__ Ssum, const float* __restrict__ Asum,
               const float* __restrict__ tx, const float* __restrict__ tc,
               float* __restrict__ ex) {
  const int z = blockIdx.z;
  const int lane = threadIdx.x & 31;
  const int wid  = threadIdx.x >> 5;
  const int m0 = blockIdx.y * 64 + (wid >> 2) * 32;
  const int c0 = (wid & 3) * 64;
  const float* txz = tx + (size_t)z * NP * H0D;
  const float* tcz = tc + (size_t)z * NP * H0D;

  v8f axc[2][4] = {}, avc[2][4] = {};
  for (int m = 0; m < NP; m += 4) {
    if (m + 16 < NP)
      __builtin_prefetch(&txz[(size_t)(m + 16) * H0D + c0 + (lane & 15)], 0, 1);
    v2f as0 = ld_a(Ssum, NP, m0,      m, lane, NP, NP);
    v2f as1 = ld_a(Ssum, NP, m0 + 16, m, lane, NP, NP);
    v2f aa0 = ld_a(Asum, NP, m0,      m, lane, NP, NP);
    v2f aa1 = ld_a(Asum, NP, m0 + 16, m, lane, NP, NP);
#pragma unroll
    for (int t = 0; t < 4; ++t) {
      v2f bx = ld_b(txz, H0D, m, c0 + 16 * t, lane, NP, H0D);
      v2f bc = ld_b(tcz, H0D, m, c0 + 16 * t, lane, NP, H0D);
      axc[0][t] = WMMA_F32(as0, bx, axc[0][t]);
      axc[1][t] = WMMA_F32(as1, bx, axc[1][t]);
      avc[0][t] = WMMA_F32(aa0, bc, avc[0][t]);
      avc[1][t] = WMMA_F32(aa1, bc, avc[1][t]);
    }
  }
  float* exz = ex + (size_t)z * NP * H0D;
#pragma unroll
  for (int r = 0; r < 2; ++r) {
#pragma unroll
    for (int t = 0; t < 4; ++t) {
      int col = c0 + 16 * t + (lane & 15);
      int rbase = m0 + 16 * r + ((lane >> 4) << 3);
#pragma unroll
      for (int j = 0; j < 8; ++j) {
        int row = rbase + j;
        exz[(size_t)row * H0D + col] = ex_relu(axc[r][t][j], avc[r][t][j]);
      }
    }
  }
}

// h[b,n,:] = sum_k gamma[b,k,n] * ex[b,k,n,:]   (padded rows -> 0)
__global__ void mix_k(const float* __restrict__ gamma, const float* __restrict__ ex,
                      float* __restrict__ h) {
  int idx = blockIdx.x * 256 + threadIdx.x;
  if (idx >= WB * NP * H0D) return;
  int hh = idx & (H0D - 1);
  int n  = (idx / H0D) & (NP - 1);
  int b  = idx / (H0D * NP);
  float acc = 0.f;
  if (n < NN) {
#pragma unroll
    for (int k = 0; k < KK; ++k) {
      size_t zr = ((size_t)b * KK + k) * NP + n;
      acc += gamma[zr] * ex[zr * H0D + hh];
    }
  }
  h[idx] = acc;
}

// ---------------------------------------------------------------------------
extern "C" void kernel_launch(void* const* d_in, const int* in_sizes, int n_in,
                              void* d_out, int out_size, void* d_ws, size_t ws_size,
                              hipStream_t stream) {
  const float* shift   = (const float*)d_in[0];  // [N,N]
  const float* feat    = (const float*)d_in[1];  // [B,N,F0] (NaN = missing)
  const float* all_A   = (const float*)d_in[2];  // [N,N]
  const float* mu      = (const float*)d_in[3];  // [K,F0]
  const float* logvars = (const float*)d_in[4];  // [K,F0]
  const float* logp    = (const float*)d_in[5];  // [K]
  const float* W0      = (const float*)d_in[6];  // [F0,H0]
  const float* W1      = (const float*)d_in[7];  // [H0,H1]
  const float* W2      = (const float*)d_in[8];  // [H1,P]
  const float* b2      = (const float*)d_in[9];  // [P]
  float* out = (float*)d_out;                    // [B,N,P]

  // workspace layout (floats)
  float* w = (float*)d_ws;
  size_t o = 0;
  const size_t MSQ = (size_t)NP * NP;
  float* Spad = w + o; o += MSQ;
  float* S2   = w + o; o += MSQ;
  float* S3   = w + o; o += MSQ;
  float* Apad = w + o; o += MSQ;
  float* A2   = w + o; o += MSQ;
  float* A3   = w + o; o += MSQ;
  float* Ssum = w + o; o += MSQ;
  float* Asum = w + o; o += MSQ;
  float* vars = w + o; o += 768;
  float* ivar = w + o; o += 768;
  float* cst  = w + o; o += 64;
  float* gam  = w + o; o += (size_t)WB * KK * NP;
  const size_t TXE = (size_t)WB * KK * NP * H0D;
  float* tx = w + o; o += TXE;
  float* tc = w + o; o += TXE;
  float* ex = w + o; o += TXE;
  const size_t HE = (size_t)WB * NP * H0D;
  float* hbuf = w + o; o += HE;
  float* tbuf = w + o; o += HE;
  float* h2   = w + o; o += HE;
  (void)ws_size; (void)n_in; (void)in_sizes; (void)out_size;

  const dim3 blk(256);

  // 1) pad shift/all_A to 512x512
  pad512_k<<<dim3((NP * NP + 255) / 256), blk, 0, stream>>>(shift, Spad);
  pad512_k<<<dim3((NP * NP + 255) / 256), blk, 0, stream>>>(all_A, Apad);

  // 2) matrix powers via WMMA GEMM (512^3 each)
  gemm_wmma_k<0><<<dim3(2, 8, 1), blk, 0, stream>>>(Spad, Spad, nullptr, S2,
      NP, NP, NP, NP, NP, NP, 0, 0, 0);
  gemm_wmma_k<0><<<dim3(2, 8, 1), blk, 0, stream>>>(S2, Spad, nullptr, S3,
      NP, NP, NP, NP, NP, NP, 0, 0, 0);
  gemm_wmma_k<0><<<dim3(2, 8, 1), blk, 0, stream>>>(Apad, Apad, nullptr, A2,
      NP, NP, NP, NP, NP, NP, 0, 0, 0);
  gemm_wmma_k<0><<<dim3(2, 8, 1), blk, 0, stream>>>(A2, Apad, nullptr, A3,
      NP, NP, NP, NP, NP, NP, 0, 0, 0);

  // 3) Ssum / Asum (includes /ORDER)
  combine_k<<<dim3((NP * NP + 255) / 256), blk, 0, stream>>>(
      Spad, S2, S3, Apad, A2, A3, Ssum, Asum);

  // 4) GMM statistics + responsibilities
  prep_k<<<dim3(1), blk, 0, stream>>>(logvars, logp, vars, ivar, cst);
  gamma_k<<<dim3((WB * NN + 255) / 256), blk, 0, stream>>>(feat, mu, ivar, cst, gam);

  // 5) imputed dual GEMM -> tx, tc   [160 batches of 512x144x256]
  impute_gemm_k<<<dim3(1, NP / 64, WB * KK), blk, 0, stream>>>(
      feat, mu, vars, W0, tx, tc);

  // 6) fused diffusion + E[relu] -> ex  [160 batches of 512x512x256]
  conv_ex_k<<<dim3(1, NP / 64, WB * KK), blk, 0, stream>>>(Ssum, Asum, tx, tc, ex);

  // 7) mixture collapse -> h [B,512,256]
  mix_k<<<dim3((WB * NP * H0D + 255) / 256), blk, 0, stream>>>(gam, ex, hbuf);

  // 8) t = h @ W1
  gemm_wmma_k<0><<<dim3(1, NP / 64, WB), blk, 0, stream>>>(hbuf, W1, nullptr, tbuf,
      NP, H0D, H1D, H0D, H1D, H1D, (long long)NP * H0D, 0, (long long)NP * H1D);

  // 9) h2 = relu(Ssum @ t)
  gemm_wmma_k<1><<<dim3(1, NP / 64, WB), blk, 0, stream>>>(Ssum, tbuf, nullptr, h2,
      NP, NP, H1D, NP, H1D, H1D, 0, (long long)NP * H1D, (long long)NP * H1D);

  // 10) out = h2 @ W2 + b2   (only n < 500 rows written)
  gemm_wmma_k<2><<<dim3(2, (NN + 63) / 64, WB), blk, 0, stream>>>(h2, W2, b2, out,
      NN, H1D, PPD, H1D, PPD, PPD, (long long)NP * H1D, 0, (long long)NN * PPD);
}